// DWT3D_18631568130940
// MI455X (gfx1250) — compile-verified
//
#include <hip/hip_runtime.h>

// 3D Haar DWT (N=4, C=4, D=64, H=256, W=256) -> 8 subbands [N,C,32,128,128].
//
// Roofline: 268 MB in + 268 MB out, ~32 flops per 2x2x2 block -> AI ~0.5
// flop/byte. HBM floor at 23.3 TB/s ~= 23 us. WMMA is intentionally absent
// (dense-matrix formulation inflates FLOPs ~64x/dim and is compute-bound
// ~20x above the memory floor). The CDNA5-specific data path used instead:
// async global->LDS DMA (ASYNCcnt) staging of the input rows, which keeps
// HBM traffic identical while deepening the memory pipeline; plus
// global_prefetch_b8 and non-temporal b128 stores.

typedef float f8 __attribute__((ext_vector_type(8)));
typedef float f4 __attribute__((ext_vector_type(4)));
typedef int   v4i __attribute__((ext_vector_type(4)));

typedef __attribute__((address_space(1))) v4i gv4i;   // global int4
typedef __attribute__((address_space(3))) v4i lv4i;   // LDS int4
typedef __attribute__((address_space(3))) char lchar;

__device__ __forceinline__ void async_copy_b128(const float* g, float* l)
{
#if __has_builtin(__builtin_amdgcn_global_load_async_to_lds_b128)
    __builtin_amdgcn_global_load_async_to_lds_b128((gv4i*)g, (lv4i*)l, 0, 0);
#else
    unsigned loff = (unsigned)(unsigned long long)(lchar*)l;  // LDS byte offset
    asm volatile("global_load_async_to_lds_b128 %0, %1, off"
                 :: "v"(loff), "v"(g) : "memory");
#endif
}

__device__ __forceinline__ void wait_async0()
{
#if __has_builtin(__builtin_amdgcn_s_wait_asynccnt)
    __builtin_amdgcn_s_wait_asynccnt(0);
#else
    asm volatile("s_wait_asynccnt 0x0" ::: "memory");
#endif
    asm volatile("" ::: "memory");  // compiler barrier: keep LDS reads below
}

__global__ __launch_bounds__(256)
void dwt3d_haar_kernel(const float* __restrict__ x, float* __restrict__ out)
{
    constexpr float S3 = 0.35355339059327379f;  // 2^-1.5 (three Haar stages)

    __shared__ float smem[8 * 1024];            // 4 KB slab per wave32, 32 KB/block

    const int lane = threadIdx.x & 31;
    const int warp = threadIdx.x >> 5;
    const int r    = (blockIdx.x << 3) + warp;  // 0 .. 65535 row-tasks
    const int h2   = r & 127;                   // output h'
    const int d2   = (r >> 7) & 31;             // output d'
    const int nc   = r >> 12;                   // fused n*C + c, 0 .. 15

    // Input element offset for (nc, 2*d2, 2*h2, lane*8); strides in floats:
    // W row = 256, H = 256, D = 65536, nc = 64*65536.
    const float* base =
        x + ((((nc << 6) + (d2 << 1)) << 8) + (h2 << 1)) * 256 + (lane << 3);

    // CDNA5 speculative prefetch (global_prefetch_b8) of future rows into L2.
    __builtin_prefetch(base + 4096, 0, 1);

    // Stage the 2x2 (d,h) row footprint into this wave's private LDS slab via
    // async DMA: 8x global_load_async_to_lds_b128 (32 B/lane/row, contiguous
    // 1 KB per row across the wave). Wave-private slab => no barrier; ASYNCcnt
    // orders completion.
    float* slab = smem + (warp << 10);
    const int lo = lane << 3;
    async_copy_b128(base,                 slab + lo);                // d0 h0
    async_copy_b128(base + 4,             slab + lo + 4);
    async_copy_b128(base + 256,           slab + 256 + lo);          // d0 h1
    async_copy_b128(base + 260,           slab + 256 + lo + 4);
    async_copy_b128(base + 65536,         slab + 512 + lo);          // d1 h0
    async_copy_b128(base + 65540,         slab + 512 + lo + 4);
    async_copy_b128(base + 65792,         slab + 768 + lo);          // d1 h1
    async_copy_b128(base + 65796,         slab + 768 + lo + 4);
    wait_async0();

    // Read back 8 floats per row (2x ds_load_b128 each, 32 B-aligned).
    f8 v00 = *(const f8*)(slab + lo);
    f8 v01 = *(const f8*)(slab + 256 + lo);
    f8 v10 = *(const f8*)(slab + 512 + lo);
    f8 v11 = *(const f8*)(slab + 768 + lo);

    // Stage 1: W butterfly (adjacent column pairs) — lane-local VALU.
    f4 wl00 = v00.even + v00.odd, wh00 = v00.even - v00.odd;
    f4 wl01 = v01.even + v01.odd, wh01 = v01.even - v01.odd;
    f4 wl10 = v10.even + v10.odd, wh10 = v10.even - v10.odd;
    f4 wl11 = v11.even + v11.odd, wh11 = v11.even - v11.odd;

    // Stage 2: H butterfly (row pairs within each d).
    f4 ll0 = wl00 + wl01, hl0 = wl00 - wl01;
    f4 lh0 = wh00 + wh01, hh0 = wh00 - wh01;
    f4 ll1 = wl10 + wl11, hl1 = wl10 - wl11;
    f4 lh1 = wh10 + wh11, hh1 = wh10 - wh11;

    // Stage 3: D butterfly + scale; subband order (D,H,W) matches the
    // reference return order (LLL, LLH, LHL, LHH, HLL, HLH, HHL, HHH).
    float* ob = out + (((nc << 5) + d2) * 128 + h2) * 128 + (lane << 2);
    constexpr int SB = 32 * 128 * 128 * 16;  // 8,388,608 floats per subband

    __builtin_nontemporal_store((ll0 + ll1) * S3, (f4*)(ob + 0 * SB)); // LLL
    __builtin_nontemporal_store((lh0 + lh1) * S3, (f4*)(ob + 1 * SB)); // LLH
    __builtin_nontemporal_store((hl0 + hl1) * S3, (f4*)(ob + 2 * SB)); // LHL
    __builtin_nontemporal_store((hh0 + hh1) * S3, (f4*)(ob + 3 * SB)); // LHH
    __builtin_nontemporal_store((ll0 - ll1) * S3, (f4*)(ob + 4 * SB)); // HLL
    __builtin_nontemporal_store((lh0 - lh1) * S3, (f4*)(ob + 5 * SB)); // HLH
    __builtin_nontemporal_store((hl0 - hl1) * S3, (f4*)(ob + 6 * SB)); // HHL
    __builtin_nontemporal_store((hh0 - hh1) * S3, (f4*)(ob + 7 * SB)); // HHH
}

extern "C" void kernel_launch(void* const* d_in, const int* in_sizes, int n_in,
                              void* d_out, int out_size, void* d_ws, size_t ws_size,
                              hipStream_t stream)
{
    (void)in_sizes; (void)n_in; (void)out_size; (void)d_ws; (void)ws_size;
    // d_in[0] = x [4,4,64,256,256] fp32. The six filter matrices (d_in[1..6])
    // are deterministic Haar constants; coefficients are folded into the
    // butterfly (signs + single 2^-1.5 scale).
    const float* x = (const float*)d_in[0];
    float* out     = (float*)d_out;

    // 65536 row-tasks (nc * d' * h'), one wave32 per task, 8 waves per block.
    dim3 grid(8192), block(256);
    hipLaunchKernelGGL(dwt3d_haar_kernel, grid, block, 0, stream, x, out);
}